// LBQ_32950989095685
// MI455X (gfx1250) — compile-verified
//
#include <hip/hip_runtime.h>

// Problem geometry (fixed by the reference): B=16, C=3, H=W=512
#define NCH              48          // B*C channels
#define HW               262144      // H*W elements per channel
#define PARTS            64          // stage-A partial blocks per channel
#define C_BLOCKS_PER_CH  16          // stage-C blocks per channel
#define C_ELEMS          16384       // elements per stage-C block
#define TILE             4096        // floats per pipeline tile (16 KB)
#define CHUNK            1024        // floats per async round (256 thr * 4)

#if __has_builtin(__builtin_amdgcn_global_load_async_to_lds_b128)
#define ASYNC_BUILTIN 1
#endif

typedef int v4i __attribute__((vector_size(16)));
typedef __attribute__((address_space(1))) v4i* gv4p;
typedef __attribute__((address_space(3))) v4i* lv4p;
typedef __attribute__((address_space(1))) void* gvoidp;
typedef __attribute__((address_space(3))) void* lvoidp;

// ---- CDNA5 async global->LDS copy of 16 bytes (one b128 per lane) ----------
__device__ __forceinline__ void async_copy16(const float* gsrc, float* ldst) {
#if defined(ASYNC_BUILTIN)
  gv4p g = (gv4p)(gvoidp)(void*)gsrc;   // drop const, addrspacecast, retype
  lv4p l = (lv4p)(lvoidp)(void*)ldst;
  __builtin_amdgcn_global_load_async_to_lds_b128(g, l, 0, 0);
#else
  unsigned laddr = (unsigned)(size_t)(lvoidp)(void*)ldst;
  unsigned long long gaddr = (unsigned long long)(size_t)gsrc;
  asm volatile("global_load_async_to_lds_b128 %0, %1, off"
               :: "v"(laddr), "v"(gaddr)
               : "memory");
#endif
}

template <int N>
__device__ __forceinline__ void wait_async() {
#if __has_builtin(__builtin_amdgcn_s_wait_asynccnt)
  __builtin_amdgcn_s_wait_asynccnt(N);
#else
  asm volatile("s_wait_asynccnt %0" :: "n"(N) : "memory");
#endif
}

// ---- first-match region search: idx = first r with L[r] <= x < L[r+1] ------
// (right[r] == left[r+1]; L[16] = max+1e-6; t0 = x>=min is always true)
__device__ __forceinline__ float quant1(float x, const float L[17]) {
  float v = L[0];
  bool tn = !(x < L[16]);            // plays the role of t_16
  #pragma unroll
  for (int r = 15; r >= 1; --r) {
    bool tr = x >= L[r];
    v = (tr && !tn) ? L[r] : v;      // descending: lowest matching r wins
    tn = tr;
  }
  v = tn ? v : L[0];                 // r = 0: mask = !t_1
  return v;
}

// ---- Stage A: per-block partial min/max (4096 elems / block) ---------------
__global__ __launch_bounds__(256) void k_minmax(const float* __restrict__ x,
                                                float2* __restrict__ part) {
  const int tid = threadIdx.x;
  const float4* p = (const float4*)x + (size_t)blockIdx.x * 1024 + tid;
  float mn = 3.4e38f, mx = -3.4e38f;
  #pragma unroll
  for (int j = 0; j < 4; ++j) {
    float4 v = p[j * 256];
    mn = fminf(mn, fminf(fminf(v.x, v.y), fminf(v.z, v.w)));
    mx = fmaxf(mx, fmaxf(fmaxf(v.x, v.y), fmaxf(v.z, v.w)));
  }
  #pragma unroll
  for (int off = 16; off >= 1; off >>= 1) {      // wave32 reduction
    mn = fminf(mn, __shfl_xor(mn, off, 32));
    mx = fmaxf(mx, __shfl_xor(mx, off, 32));
  }
  __shared__ float smn[8], smx[8];
  const int wid = tid >> 5, lane = tid & 31;
  if (lane == 0) { smn[wid] = mn; smx[wid] = mx; }
  __syncthreads();
  if (tid == 0) {
    float a = smn[0], b = smx[0];
    #pragma unroll
    for (int i = 1; i < 8; ++i) { a = fminf(a, smn[i]); b = fmaxf(b, smx[i]); }
    part[blockIdx.x] = make_float2(a, b);
  }
}

// ---- Stage B: final reduce + build 17-entry threshold table per channel ----
__global__ __launch_bounds__(64) void k_tables(const float2* __restrict__ part,
                                               const float* __restrict__ rp,
                                               float* __restrict__ tab) {
  const int c = blockIdx.x, t = threadIdx.x;     // 64 threads = 64 partials
  float2 pv = part[c * PARTS + t];
  float mn = pv.x, mx = pv.y;
  #pragma unroll
  for (int off = 16; off >= 1; off >>= 1) {
    mn = fminf(mn, __shfl_xor(mn, off, 32));
    mx = fmaxf(mx, __shfl_xor(mx, off, 32));
  }
  __shared__ float smn[2], smx[2];
  __shared__ float bmn, bmx;
  if ((t & 31) == 0) { smn[t >> 5] = mn; smx[t >> 5] = mx; }
  __syncthreads();
  if (t == 0) { bmn = fminf(smn[0], smn[1]); bmx = fmaxf(smx[0], smx[1]); }
  __syncthreads();
  if (t < 17) {
    const float mnv = bmn, mxv = bmx;
    float L;
    if (t == 0)      L = mnv;
    else if (t < 16) L = rp[c * 15 + (t - 1)] * (mxv - mnv) + mnv;
    else             L = mxv + 1e-6f;            // right[15], f32 like the ref
    tab[c * 32 + t] = L;
  }
}

// ---- Stage C: async double-buffered stream + quantize ----------------------
__global__ __launch_bounds__(256) void k_quant(const float* __restrict__ x,
                                               const float* __restrict__ tab,
                                               float* __restrict__ out) {
  __shared__ float lds[2 * TILE];                // 32 KB of the 320 KB WGP LDS
  const int tid = threadIdx.x;
  const int ch = blockIdx.x >> 4;                // 16 blocks per channel
  const size_t base = (size_t)blockIdx.x * C_ELEMS;

  float L[17];
  const float* tb = tab + ch * 32;               // block-uniform -> s_loads
  #pragma unroll
  for (int i = 0; i < 17; ++i) L[i] = tb[i];

  auto issue = [&](int t, int buf) {             // 4 x b128 per thread = 1 tile
    const float* src = x + base + (size_t)t * TILE + tid * 4;
    float* dst = &lds[buf * TILE + tid * 4];
    #pragma unroll
    for (int j = 0; j < 4; ++j)
      async_copy16(src + j * CHUNK, dst + j * CHUNK);
  };

  issue(0, 0);
  #pragma unroll
  for (int t = 0; t < 4; ++t) {
    if (t < 3) issue(t + 1, (t + 1) & 1);        // prefetch next tile
    if (t < 3) wait_async<4>();                  // tile t done (in-order loads)
    else       wait_async<0>();
    // each thread consumes exactly the bytes it issued -> no barrier needed
    const float* lb = &lds[(t & 1) * TILE + tid * 4];
    float* o = out + base + (size_t)t * TILE + tid * 4;
    #pragma unroll
    for (int j = 0; j < 4; ++j) {
      float4 v = *(const float4*)(lb + j * CHUNK);
      float4 q;
      q.x = quant1(v.x, L);
      q.y = quant1(v.y, L);
      q.z = quant1(v.z, L);
      q.w = quant1(v.w, L);
      *(float4*)(o + j * CHUNK) = q;
    }
  }
}

extern "C" void kernel_launch(void* const* d_in, const int* in_sizes, int n_in,
                              void* d_out, int out_size, void* d_ws, size_t ws_size,
                              hipStream_t stream) {
  const float* x  = (const float*)d_in[0];   // (16,3,512,512) f32
  const float* rp = (const float*)d_in[1];   // (48,15) f32
  float* out = (float*)d_out;
  float* ws  = (float*)d_ws;

  float2* part = (float2*)ws;                // 3072 float2  (24 KB)
  float*  tabv = ws + 8192;                  // 48*32 floats ( 6 KB)

  k_minmax<<<NCH * PARTS, 256, 0, stream>>>(x, part);
  k_tables<<<NCH, 64, 0, stream>>>(part, rp, tabv);
  k_quant <<<NCH * C_BLOCKS_PER_CH, 256, 0, stream>>>(x, tabv, out);
}